// TemporalLoss_12661563589026
// MI455X (gfx1250) — compile-verified
//
#include <hip/hip_runtime.h>
#include <math.h>
#include <stdint.h>

// Banded DTW (Sakoe-Chiba w=1) as an ordered reduction of 3x3 min-plus matrices.
// Row step:  n0 = c0 + min(p0,p1); n1 = c1 + min(n0,p1,p2); n2 = c2 + min(n1,p2)
// is min-plus-affine => whole scan = M_N (x) ... (x) M_1 in the (min,+) semiring,
// which is associative -> parallel tree reduction (order preserved).
// Final answer = M_total[1][1]  (initial carry = (INF, 0, INF)).
//
// CDNA5 specifics used here:
//  - GLOBAL_LOAD_ASYNC_TO_LDS_B128 for bulk global->LDS staging (ASYNCcnt path,
//    no VGPR round-trip), completed with s_wait_asynccnt + workgroup barrier.

#define N_SEQ 131072
#define BLOCKS 64
#define THREADS 256
#define ROWS_PER_BLOCK (N_SEQ / BLOCKS)             // 2048
#define ROWS_PER_THREAD (ROWS_PER_BLOCK / THREADS)  // 8
#define VEC4_PER_THREAD ((ROWS_PER_BLOCK / 4) / THREADS)  // 2

// out = B (x) A  in (min,+):  out[i][j] = min_k ( B[i][k] + A[k][j] )
// A is the EARLIER segment, B the LATER one (B applied after A).
__device__ __forceinline__ void mp_combine(const float* __restrict__ A,
                                           const float* __restrict__ B,
                                           float* __restrict__ out) {
#pragma unroll
    for (int i = 0; i < 3; ++i) {
#pragma unroll
        for (int j = 0; j < 3; ++j) {
            float v = B[i * 3 + 0] + A[0 * 3 + j];
            v = fminf(v, B[i * 3 + 1] + A[1 * 3 + j]);
            v = fminf(v, B[i * 3 + 2] + A[2 * 3 + j]);
            out[i * 3 + j] = v;
        }
    }
}

// One async 16B/lane global->LDS transfer. ldsOff is the LDS byte offset
// (low 32 bits of a generic pointer to __shared__), gaddr a 16B-aligned
// global address. Tracked with ASYNCcnt.
__device__ __forceinline__ void async_copy_b128(unsigned ldsOff, const float* gaddr) {
    asm volatile("global_load_async_to_lds_b128 %0, %1, off"
                 :
                 : "v"(ldsOff), "v"(gaddr)
                 : "memory");
}

__device__ __forceinline__ void wait_asynccnt0() {
    asm volatile("s_wait_asynccnt 0x0" ::: "memory");
}

__global__ void __launch_bounds__(THREADS)
dtw_partial(const float* __restrict__ out_seq,
            const float* __restrict__ tgt_seq,
            float* __restrict__ block_mats) {
    __shared__ float s_out[ROWS_PER_BLOCK];
    // target interior lives at s_tgt[4 .. 4+ROWS_PER_BLOCK) so it is 16B aligned
    // for b128 async transfers; the two clamped edge cells sit at [3] and [4+ROWS].
    __shared__ float s_tgt[ROWS_PER_BLOCK + 8];
    __shared__ float s_red[THREADS * 9];

    const int tid  = threadIdx.x;
    const int base = blockIdx.x * ROWS_PER_BLOCK;  // covers rows i = base+1 .. base+ROWS_PER_BLOCK

    // ---- async global->LDS staging (2 x b128 per array per thread) ----
    const unsigned lds_out = (unsigned)(uintptr_t)&s_out[0];
    const unsigned lds_tgt = (unsigned)(uintptr_t)&s_tgt[4];
    const float* g_out = out_seq + base;
    const float* g_tgt = tgt_seq + base;
#pragma unroll
    for (int t = 0; t < VEC4_PER_THREAD; ++t) {
        const int v = tid + t * THREADS;  // float4 index within the block's tile
        async_copy_b128(lds_out + 16u * (unsigned)v, g_out + 4 * v);
        async_copy_b128(lds_tgt + 16u * (unsigned)v, g_tgt + 4 * v);
    }
    // clamped edge cells (feed only band entries that are forced to INF below)
    if (tid == 0) {
        int gl = base - 1;               gl = gl < 0 ? 0 : gl;
        int gh = base + ROWS_PER_BLOCK;  gh = gh > N_SEQ - 1 ? N_SEQ - 1 : gh;
        s_tgt[3] = tgt_seq[gl];
        s_tgt[4 + ROWS_PER_BLOCK] = tgt_seq[gh];
    }
    wait_asynccnt0();   // this wave's async transfers done writing LDS
    __syncthreads();    // all waves' transfers + edge stores visible

    const float INF = __builtin_inff();
    // Min-plus identity matrix; kept as 3 column vectors (m0j,m1j,m2j); the row
    // step is applied to each column (21 ops/row vs 45 for a full 3x3 product).
    float m00 = 0.f, m10 = INF, m20 = INF;   // column 0
    float m01 = INF, m11 = 0.f, m21 = INF;   // column 1
    float m02 = INF, m12 = INF, m22 = 0.f;   // column 2

    const int r0 = tid * ROWS_PER_THREAD;    // this thread's local row offset
#pragma unroll
    for (int r = 0; r < ROWS_PER_THREAD; ++r) {
        const int lr = r0 + r;               // local index; global row i = base + lr + 1
        const int i  = base + lr + 1;
        const float o  = s_out[lr];
        const float a0 = s_tgt[lr + 3];      // target[i-2] (clamped at i=1)
        const float a1 = s_tgt[lr + 4];      // target[i-1]
        const float a2 = s_tgt[lr + 5];      // target[i]   (clamped at i=N)
        const float c0 = (i >= 2)         ? fabsf(o - a0) : INF;  // j = i-1 edge
        const float c1 = fabsf(o - a1);                           // j = i
        const float c2 = (i <= N_SEQ - 1) ? fabsf(o - a2) : INF;  // j = i+1 edge

        float n0, n1, n2;
        n0 = c0 + fminf(m00, m10);
        n1 = c1 + fminf(n0, fminf(m10, m20));
        n2 = c2 + fminf(n1, m20);
        m00 = n0; m10 = n1; m20 = n2;

        n0 = c0 + fminf(m01, m11);
        n1 = c1 + fminf(n0, fminf(m11, m21));
        n2 = c2 + fminf(n1, m21);
        m01 = n0; m11 = n1; m21 = n2;

        n0 = c0 + fminf(m02, m12);
        n1 = c1 + fminf(n0, fminf(m12, m22));
        n2 = c2 + fminf(n1, m22);
        m02 = n0; m12 = n1; m22 = n2;
    }

    float* my = &s_red[tid * 9];
    my[0] = m00; my[1] = m01; my[2] = m02;
    my[3] = m10; my[4] = m11; my[5] = m12;
    my[6] = m20; my[7] = m21; my[8] = m22;
    __syncthreads();

    // Order-preserving adjacent-pair tree: thread tid (segment earlier) absorbs
    // tid+s (the segment immediately after it) when tid % (2s) == 0.
    for (int s = 1; s < THREADS; s <<= 1) {
        if ((tid & (2 * s - 1)) == 0) {
            float A[9], B[9], C[9];
#pragma unroll
            for (int q = 0; q < 9; ++q) {
                A[q] = s_red[tid * 9 + q];        // earlier
                B[q] = s_red[(tid + s) * 9 + q];  // later
            }
            mp_combine(A, B, C);
#pragma unroll
            for (int q = 0; q < 9; ++q) s_red[tid * 9 + q] = C[q];
        }
        __syncthreads();
    }

    if (tid == 0) {
#pragma unroll
        for (int q = 0; q < 9; ++q) block_mats[blockIdx.x * 9 + q] = s_red[q];
    }
}

__global__ void __launch_bounds__(BLOCKS)
dtw_final(const float* __restrict__ block_mats, float* __restrict__ result) {
    __shared__ float s_red[BLOCKS * 9];
    const int tid = threadIdx.x;
#pragma unroll
    for (int q = 0; q < 9; ++q) s_red[tid * 9 + q] = block_mats[tid * 9 + q];
    __syncthreads();

    for (int s = 1; s < BLOCKS; s <<= 1) {
        if ((tid & (2 * s - 1)) == 0) {
            float A[9], B[9], C[9];
#pragma unroll
            for (int q = 0; q < 9; ++q) {
                A[q] = s_red[tid * 9 + q];        // earlier
                B[q] = s_red[(tid + s) * 9 + q];  // later
            }
            mp_combine(A, B, C);
#pragma unroll
            for (int q = 0; q < 9; ++q) s_red[tid * 9 + q] = C[q];
        }
        __syncthreads();
    }

    if (tid == 0) {
        // answer = (M_total (x) (INF,0,INF))[1] = M_total[1][1]
        result[0] = s_red[1 * 3 + 1];
    }
}

extern "C" void kernel_launch(void* const* d_in, const int* in_sizes, int n_in,
                              void* d_out, int out_size, void* d_ws, size_t ws_size,
                              hipStream_t stream) {
    (void)in_sizes; (void)n_in; (void)out_size; (void)ws_size;
    const float* out_seq = (const float*)d_in[0];  // "output" sequence
    const float* tgt_seq = (const float*)d_in[1];  // "target" sequence
    float* block_mats = (float*)d_ws;              // BLOCKS * 9 floats scratch
    float* result = (float*)d_out;                 // single f32

    dtw_partial<<<BLOCKS, THREADS, 0, stream>>>(out_seq, tgt_seq, block_mats);
    dtw_final<<<1, BLOCKS, 0, stream>>>(block_mats, result);
}